// SpectralConv2d_39548058862299
// MI455X (gfx1250) — compile-verified
//
#include <hip/hip_runtime.h>

// SpectralConv2d (Hartley-ish) reduced to:
//   A) Y[b,i,m',n'] = sum_k H[n',k] * x[b,i,m(m'),k]      (32768 x 256 x 32 GEMM, WMMA f32 16x16x4)
//   B) Z[b,o,m',y]  = 0.5*sum_i [Y*(w+wn) + Yn*(w-wn)]    (tiny, VALU)
//   C) out[b,o,m(m'),n] = (1/65536)*sum_{y<32} H[n,y]*Z   (32768 x 32 x 256 GEMM, WMMA f32 16x16x4)
//   rows m in [32,224) of the output are exactly zero -> float4 zero-fill kernel.
// Memory-bound (~200 MB moved); fp32 kept end-to-end for accuracy.
// v2: stage A fuses both 16-col tiles per wave (x read once, 2 WMMA chains);
//     stage C sweeps 4 col-tiles per wave with a preloaded K=32 A-fragment
//     (Z L2 traffic /4, 4 interleaved WMMA chains).

typedef __attribute__((ext_vector_type(2))) float v2f;
typedef __attribute__((ext_vector_type(8))) float v8f;

// ---------------------------------------------------------------------------
// Kernel 0: build the 256x256 transform matrix H by radix-2 doubling.
// H_N[n,2j]   = H_{N/2}[n,j]                      (n <  N/2)
// H_N[n,2j+1] = cos(2*pi*n/N)    * H_{N/2}[n,j]   (n <  N/2)
// H_N[n,2j]   = 0                                 (n >= N/2)
// H_N[n,2j+1] = -sin(2*pi*m/N)   * H_{N/2}[m,j]   (n = N/2+m)
// 8 doublings starting from H_1=[1]; ends in bufA.
// ---------------------------------------------------------------------------
__global__ __launch_bounds__(256) void build_h_kernel(float* bufA, float* bufB) {
    const int tid = threadIdx.x;
    if (tid == 0) bufA[0] = 1.0f;
    __syncthreads();
    float* cur = bufA;
    float* nxt = bufB;
    for (int N = 2; N <= 256; N <<= 1) {
        const int h = N >> 1;
        const int total = N * N;
        const float w = 6.28318530717958647692f / (float)N;
        for (int idx = tid; idx < total; idx += 256) {
            const int n = idx / N;
            const int k = idx - n * N;
            const int j = k >> 1;
            float v;
            if ((k & 1) == 0) {
                v = (n < h) ? cur[n * h + j] : 0.0f;
            } else if (n < h) {
                v = cosf(w * (float)n) * cur[n * h + j];
            } else {
                const int m = n - h;
                v = -sinf(w * (float)m) * cur[m * h + j];
            }
            nxt[idx] = v;
        }
        __syncthreads();
        float* t = cur; cur = nxt; nxt = t;
    }
    // 8 swaps -> result resides in bufA
}

// ---------------------------------------------------------------------------
// Kernel 1 (stage A): Y = X_sel * H_A^T via v_wmma_f32_16x16x4_f32.
// Global row r = (b*64+i)*64 + m'; m = m' (<32) or 192+m' (>=32).
// One wave per 16-row stripe: both 16-col tiles at once (x loaded ONCE),
// two independent accumulator chains interleave in the WMMA pipe.
// ---------------------------------------------------------------------------
__global__ __launch_bounds__(256) void dht_x_kernel(const float* __restrict__ x,
                                                    const float* __restrict__ H,
                                                    float* __restrict__ Y) {
    const int wave = (blockIdx.x << 3) | (threadIdx.x >> 5);  // 0..2047 (row tile)
    const int lane = threadIdx.x & 31;
    const int rowbase = wave << 4;
    const int lmod = lane & 15;
    const int kh   = lane >> 4;  // K half: lanes 0-15 -> K{0,1}, 16-31 -> K{2,3}

    const int r  = rowbase + lmod;
    const int mp = r & 63;
    const int bi = r >> 6;
    const int m  = (mp < 32) ? mp : (192 + mp);
    const float* xrow  = x + ((size_t)bi * 256 + (size_t)m) * 256;
    const float* hrow0 = H + (size_t)(lmod) * 256;        // n' = 0..15
    const float* hrow1 = H + (size_t)(16 + lmod) * 256;   // n' = 16..31

    v8f c0 = {0.f, 0.f, 0.f, 0.f, 0.f, 0.f, 0.f, 0.f};
    v8f c1 = {0.f, 0.f, 0.f, 0.f, 0.f, 0.f, 0.f, 0.f};
    for (int k0 = 0; k0 < 256; k0 += 4) {
        const int ko = k0 + 2 * kh;
        v2f a  = *(const v2f*)(xrow + ko);
        v2f b0 = *(const v2f*)(hrow0 + ko);
        v2f b1 = *(const v2f*)(hrow1 + ko);
        c0 = __builtin_amdgcn_wmma_f32_16x16x4_f32(false, a, false, b0,
                                                   (short)0, c0, false, false);
        c1 = __builtin_amdgcn_wmma_f32_16x16x4_f32(false, a, false, b1,
                                                   (short)0, c1, false, false);
    }

    // C/D layout: VGPR rr -> M = rr + 8*(lane/16), N = lane%16
    float* y0 = Y + (size_t)lmod;
    float* y1 = Y + (size_t)(16 + lmod);
#pragma unroll
    for (int rr = 0; rr < 8; ++rr) {
        const size_t row = (size_t)(rowbase + rr + (kh << 3)) * 32;
        y0[row] = c0[rr];
        y1[row] = c1[rr];
    }
}

// ---------------------------------------------------------------------------
// Kernel 2 (stage B): per mode pair (m',y):
//   Z[b,o] = 0.5 * sum_i [ Y[b,i,m',y]*(w+wn) + Y[b,i,nm',ny]*(w-wn) ]
// where wn uses negated-mod-32 indices inside the 32x32 block.
// 2048 blocks (one per mode pair) x 512 threads (b,o).
// ---------------------------------------------------------------------------
__global__ __launch_bounds__(512) void mode_mul_kernel(const float* __restrict__ Y,
                                                       const float* __restrict__ w1,
                                                       const float* __restrict__ w2,
                                                       float* __restrict__ Z) {
    const int mp = blockIdx.x >> 5;   // 0..63
    const int y  = blockIdx.x & 31;   // 0..31
    const int t  = threadIdx.x;
    const int b  = t >> 6;            // 0..7
    const int o  = t & 63;            // 0..63

    const int xm = mp & 31;
    const int nx = (32 - xm) & 31;
    const int ny = (32 - y) & 31;
    const int nmp = (mp < 32) ? nx : (32 + nx);
    const float* W = (mp < 32) ? w1 : w2;

    float acc = 0.0f;
    for (int i = 0; i < 64; ++i) {
        const float A1 = Y[((size_t)(b * 64 + i) * 64 + mp) * 32 + y];
        const float A2 = Y[((size_t)(b * 64 + i) * 64 + nmp) * 32 + ny];
        const float wv  = W[((size_t)(i * 64 + o) * 32 + xm) * 32 + y];
        const float wnv = W[((size_t)(i * 64 + o) * 32 + nx) * 32 + ny];
        acc += A1 * (wv + wnv) + A2 * (wv - wnv);
    }
    Z[((size_t)(b * 64 + o) * 64 + mp) * 32 + y] = 0.5f * acc;
}

// ---------------------------------------------------------------------------
// Kernel 3: zero-fill output rows m in [32,224)  (exactly zero in the math).
// 8*64*192*256 floats = 6291456 float4 stores.
// ---------------------------------------------------------------------------
__global__ __launch_bounds__(256) void zero_mid_kernel(float* __restrict__ out) {
    const size_t idx = (size_t)blockIdx.x * 256 + threadIdx.x;
    const int n4  = (int)(idx & 63);           // 64 float4 per N-row
    const size_t rest = idx >> 6;
    const int mrel = (int)(rest % 192);
    const int bo   = (int)(rest / 192);
    const float4 z = {0.f, 0.f, 0.f, 0.f};
    *(float4*)(out + ((size_t)bo * 256 + 32 + (size_t)mrel) * 256 + (size_t)n4 * 4) = z;
}

// ---------------------------------------------------------------------------
// Kernel 4 (stage C): out = (Z * H[:, :32]^T) / 65536 on the 64 selected rows.
// Wave = (row tile, col group of 64). The full K=32 A-fragment (8 x v2f) is
// preloaded once and reused for 4 col tiles; 4 accumulator chains interleave
// per k-step so the WMMA pipe never stalls on a single-chain RAW.
// ---------------------------------------------------------------------------
__global__ __launch_bounds__(256) void idht_kernel(const float* __restrict__ Z,
                                                   const float* __restrict__ H,
                                                   float* __restrict__ out) {
    const int wave = (blockIdx.x << 3) | (threadIdx.x >> 5);  // 0..8191
    const int lane = threadIdx.x & 31;
    const int colgrp  = wave & 3;          // 4 col groups of 64 columns
    const int rowtile = wave >> 2;         // 2048 row tiles
    const int rowbase = rowtile << 4;
    const int ncolg   = colgrp << 6;
    const int lmod    = lane & 15;
    const int kh      = lane >> 4;

    const float* zrow = Z + (size_t)(rowbase + lmod) * 32;

    // Preload the whole K=32 A-fragment for this row tile.
    v2f a[8];
#pragma unroll
    for (int j = 0; j < 8; ++j)
        a[j] = *(const v2f*)(zrow + 4 * j + 2 * kh);

    // B[k][n] = H[ncol+n][k], k<32 ; one H row pointer per col tile.
    const float* hrow[4];
#pragma unroll
    for (int ct = 0; ct < 4; ++ct)
        hrow[ct] = H + (size_t)(ncolg + (ct << 4) + lmod) * 256;

    v8f c[4];
#pragma unroll
    for (int ct = 0; ct < 4; ++ct)
        c[ct] = (v8f){0.f, 0.f, 0.f, 0.f, 0.f, 0.f, 0.f, 0.f};

#pragma unroll
    for (int j = 0; j < 8; ++j) {
        const int ko = 4 * j + 2 * kh;
#pragma unroll
        for (int ct = 0; ct < 4; ++ct) {
            v2f b = *(const v2f*)(hrow[ct] + ko);
            c[ct] = __builtin_amdgcn_wmma_f32_16x16x4_f32(false, a[j], false, b,
                                                          (short)0, c[ct], false, false);
        }
    }

    const float scale = 1.0f / 65536.0f;   // 1/(M*N)
#pragma unroll
    for (int rr = 0; rr < 8; ++rr) {
        const int r  = rowbase + rr + (kh << 3);
        const int mp = r & 63;
        const int bo = r >> 6;
        const int m  = (mp < 32) ? mp : (192 + mp);
        float* orow = out + ((size_t)bo * 256 + (size_t)m) * 256 + ncolg + lmod;
#pragma unroll
        for (int ct = 0; ct < 4; ++ct)
            orow[ct << 4] = c[ct][rr] * scale;
    }
}

// ---------------------------------------------------------------------------
extern "C" void kernel_launch(void* const* d_in, const int* in_sizes, int n_in,
                              void* d_out, int out_size, void* d_ws, size_t ws_size,
                              hipStream_t stream) {
    const float* x  = (const float*)d_in[0];  // (8,64,256,256)
    const float* w1 = (const float*)d_in[1];  // (64,64,32,32)
    const float* w2 = (const float*)d_in[2];  // (64,64,32,32)
    float* out = (float*)d_out;               // (8,64,256,256)

    char* ws = (char*)d_ws;
    float* H    = (float*)(ws);                                   // 256 KB
    float* Htmp = (float*)(ws + 256 * 1024);                      // 256 KB
    float* Y    = (float*)(ws + 512 * 1024);                      // 4 MB  (32768 x 32)
    float* Z    = (float*)(ws + 512 * 1024 + 4 * 1024 * 1024ull); // 4 MB  (32768 x 32)

    build_h_kernel<<<1, 256, 0, stream>>>(H, Htmp);
    dht_x_kernel<<<256, 256, 0, stream>>>(x, H, Y);          // 2048 waves, x read once
    mode_mul_kernel<<<2048, 512, 0, stream>>>(Y, w1, w2, Z); // one block / mode pair
    zero_mid_kernel<<<24576, 256, 0, stream>>>(out);         // zero rows m in [32,224)
    idht_kernel<<<1024, 256, 0, stream>>>(Z, H, out);        // 8192 waves, 4 chains each
}